// CoincidenceDetector_83468394430522
// MI455X (gfx1250) — compile-verified
//
#include <hip/hip_runtime.h>
#include <math.h>

// Problem dims (fixed by reference)
#define BATCH 64
#define NPAT  16384
#define DDIM  256

// Tiling
#define BTILE 16
#define NTILE 64
#define DK    32
#define NCHUNK (DDIM / DK)            // 8
#define ASYNC_PER_CHUNK 5             // 4 Ep b128 + 1 Eq b128 per thread per chunk

#define INV_TAU 0.33333334f           // 1/3 in f32
#define GATE_C  0.18887560283756183f  // exp(-5/3)

// ---------------- CDNA5 async global->LDS support (guarded) ----------------
#if defined(__has_builtin)
#if __has_builtin(__builtin_amdgcn_global_load_async_to_lds_b128) && \
    __has_builtin(__builtin_amdgcn_s_wait_asynccnt)
#define USE_ASYNC 1
#endif
#endif

#define AS1 __attribute__((address_space(1)))
#define AS3 __attribute__((address_space(3)))

typedef int v4i __attribute__((ext_vector_type(4)));
typedef float vf2 __attribute__((ext_vector_type(2)));

__device__ __forceinline__ void async_copy16(const void* gptr, void* lptr) {
#ifdef USE_ASYNC
  __builtin_amdgcn_global_load_async_to_lds_b128(
      (AS1 v4i*)gptr, (AS3 v4i*)lptr, 0, 0);
#else
  *(float4*)lptr = *(const float4*)gptr;   // synchronous fallback
#endif
}

__device__ __forceinline__ void wait_async_chunk() {
#ifdef USE_ASYNC
  __builtin_amdgcn_s_wait_asynccnt(ASYNC_PER_CHUNK);
#endif
}

__device__ __forceinline__ void wait_async_all() {
#ifdef USE_ASYNC
  __builtin_amdgcn_s_wait_asynccnt(0);
#endif
}

// ---------------- Main kernel (first in file so disasm snippet shows it) ----------------
// S[b,n] = sum_d |w_d| * gate(min(EqP*EpN, EqN*EpP)),  gate(m) = m > exp(-5/3) ? m : 0
__global__ __launch_bounds__(256)
void cd_main_kernel(const float2* __restrict__ EqT, const float2* __restrict__ EpT,
                    const float* __restrict__ w, float* __restrict__ out) {
  extern __shared__ char smem_raw[];
  float2* epA = (float2*)smem_raw;                 // DK*NTILE float2 = 16KB
  float2* epB = epA + DK * NTILE;                  // 16KB
  float2* eqA = epB + DK * NTILE;                  // DK*BTILE float2 = 4KB
  float2* eqB = eqA + DK * BTILE;                  // 4KB
  float*  wb  = (float*)(eqB + DK * BTILE);        // DDIM floats = 1KB

  const int tid = threadIdx.x;
  const int n0  = blockIdx.x * NTILE;
  const int b0  = blockIdx.y * BTILE;

  // all weights once per block (consumed only after first barrier)
  wb[tid] = fabsf(w[tid]);

  auto stage = [&](int chunk, float2* epb, float2* eqb) {
    const int d0 = chunk * DK;
    // Ep tile: DK x NTILE float2, b128 per lane, one dd-row per wave (coalesced 512B)
#pragma unroll
    for (int i = 0; i < (DK * NTILE / 2) / 256; ++i) {     // 4 iters
      int k  = i * 256 + tid;
      int dd = k / (NTILE / 2);
      int nn = (k % (NTILE / 2)) * 2;
      async_copy16(&EpT[(size_t)(d0 + dd) * NPAT + (n0 + nn)], &epb[dd * NTILE + nn]);
    }
    // Eq tile: DK x BTILE float2, 1 b128 per lane
    {
      int k  = tid;
      int dd = k / (BTILE / 2);
      int bb = (k % (BTILE / 2)) * 2;
      async_copy16(&EqT[(size_t)(d0 + dd) * BATCH + (b0 + bb)], &eqb[dd * BTILE + bb]);
    }
  };

  const int bl = (tid >> 5) * 2;   // 8 b-groups of 2
  const int nl = (tid & 31) * 2;   // 32 n-groups of 2
  float acc[2][2] = {{0.0f, 0.0f}, {0.0f, 0.0f}};

  auto compute = [&](int chunk, const float2* epb, const float2* eqb) {
    const int d0 = chunk * DK;
#pragma unroll 4
    for (int dd = 0; dd < DK; ++dd) {
      // eq read: broadcast within wave (same address for all 32 lanes)
      float4 eqv = *(const float4*)&eqb[dd * BTILE + bl];  // {EqP0,EqN0,EqP1,EqN1}
      float4 epv = *(const float4*)&epb[dd * NTILE + nl];  // {EpP0,EpN0,EpP1,EpN1}
      float wv   = wb[d0 + dd];
      // packed operands: eq2[i] = {EqP,EqN}, ep2s[j] = {EpN,EpP} (swapped)
      vf2 eq2[2]  = {{eqv.x, eqv.y}, {eqv.z, eqv.w}};
      vf2 ep2s[2] = {{epv.y, epv.x}, {epv.w, epv.z}};
#pragma unroll
      for (int i = 0; i < 2; ++i) {
#pragma unroll
        for (int j = 0; j < 2; ++j) {
          vf2 ab = eq2[i] * ep2s[j];            // v_pk_mul_f32: {e^{dt/tau}, e^{-dt/tau}}
          float m = fminf(ab.x, ab.y);          // e^{-|dt|/tau}
          float g = (m > GATE_C) ? m : 0.0f;    // window |dt| < 5
          acc[i][j] = fmaf(wv, g, acc[i][j]);
        }
      }
    }
  };

  // software pipeline, double buffered
  stage(0, epA, eqA);
  if (NCHUNK > 1) stage(1, epB, eqB);
  for (int k = 0; k < NCHUNK; ++k) {
    if (k < NCHUNK - 1) wait_async_chunk();  // own chunk k done; chunk k+1 may fly
    else                wait_async_all();
    __syncthreads();                          // whole buffer valid for all waves
    float2* ep = (k & 1) ? epB : epA;
    float2* eq = (k & 1) ? eqB : eqA;
    compute(k, ep, eq);
    __syncthreads();                          // everyone done reading before re-stage
    if (k + 2 < NCHUNK) stage(k + 2, ep, eq);
  }

  // write 2x2 outputs: coalesced float2 per row
#pragma unroll
  for (int i = 0; i < 2; ++i) {
    float2 r = make_float2(acc[i][0], acc[i][1]);
    *(float2*)&out[(size_t)(b0 + bl + i) * NPAT + (n0 + nl)] = r;
  }
}

// ---------------- Precompute: Eq[d][b] = {exp(+q/3), exp(-q/3)} ----------------
__global__ void cd_eq_kernel(const float* __restrict__ q, float2* __restrict__ EqT) {
  int id = blockIdx.x * blockDim.x + threadIdx.x;  // 0 .. BATCH*DDIM-1
  int b  = id & (BATCH - 1);
  int d  = id >> 6;                                // / BATCH
  float t  = q[b * DDIM + d];
  float a  = t * INV_TAU;
  EqT[(size_t)d * BATCH + b] = make_float2(expf(a), expf(-a));
}

// ---------------- Precompute: Ep[d][n] = {exp(+pt/3), exp(-pt/3)} (transposed) ----------------
__global__ void cd_ep_kernel(const float* __restrict__ patterns, float2* __restrict__ EpT) {
  __shared__ float2 tile[16][17];                  // padded: no bank conflicts
  int d0 = blockIdx.x * 16;
  int n0 = blockIdx.y * 16;
  int tx = threadIdx.x, ty = threadIdx.y;
  float p  = patterns[(size_t)(n0 + ty) * DDIM + (d0 + tx)];
  float s  = 1.0f / (1.0f + expf(-p));             // sigmoid
  float pt = 20.0f - 15.0f * s;                    // pattern spike time
  float a  = pt * INV_TAU;
  tile[ty][tx] = make_float2(expf(a), expf(-a));
  __syncthreads();
  // write (d = d0+ty, n = n0+tx) <- tile[tx][ty]; coalesced over tx
  EpT[(size_t)(d0 + ty) * NPAT + (n0 + tx)] = tile[tx][ty];
}

// ---------------- Launch ----------------
extern "C" void kernel_launch(void* const* d_in, const int* in_sizes, int n_in,
                              void* d_out, int out_size, void* d_ws, size_t ws_size,
                              hipStream_t stream) {
  const float* q        = (const float*)d_in[0];   // (64, 256)
  const float* patterns = (const float*)d_in[1];   // (16384, 256)
  const float* w        = (const float*)d_in[2];   // (256,)
  float* out = (float*)d_out;                      // (64, 16384)

  // workspace: EqT (128KB) at 0, EpT (32MB) at 256KB
  float2* EqT = (float2*)d_ws;
  float2* EpT = (float2*)((char*)d_ws + (256u * 1024u));

  cd_eq_kernel<<<(BATCH * DDIM) / 256, 256, 0, stream>>>(q, EqT);

  dim3 tb(16, 16), tg(DDIM / 16, NPAT / 16);
  cd_ep_kernel<<<tg, tb, 0, stream>>>(patterns, EpT);

  size_t shmem = (size_t)(2 * DK * NTILE + 2 * DK * BTILE) * sizeof(float2)
               + (size_t)DDIM * sizeof(float);     // 41KB
  dim3 g(NPAT / NTILE, BATCH / BTILE);             // 256 x 4 = 1024 blocks
  cd_main_kernel<<<g, 256, shmem, stream>>>(EqT, EpT, w, out);
}